// CBModel_46926812676771
// MI455X (gfx1250) — compile-verified
//
#include <hip/hip_runtime.h>

// Problem constants (from reference setup_inputs / CBModel)
#define B_ 32
#define C_ 18
#define H_ 256
#define W_ 256
#define PLANE (H_ * W_)                        // 65536 floats per heatmap
#define GEN_TOTAL (2ull * B_ * C_ * PLANE)     // gen_poses floats  = 75,497,472
#define STEP_TOTAL (2ull * B_ * C_ * PLANE)    // step_poses floats = 75,497,472
#define TOTAL_FLOATS (GEN_TOTAL + STEP_TOTAL)  // 150,994,944
#define TOTAL_BYTES (TOTAL_FLOATS * 4ull)      // 603,979,776 bytes

// Fill-kernel geometry: 2304 blocks * 8 waves = 18432 waves.
// Each wave issues 64 async B128 stores (512 B each):
// 18432 * 64 * 512 = 603,979,776 bytes exactly.
#define FILL_BLOCKS 2304
#define FILL_THREADS 256
#define TOTAL_WAVES (FILL_BLOCKS * (FILL_THREADS / 32))   // 18432
#define ISSUES_PER_WAVE 64u
#define BYTES_PER_ISSUE 512u
#define WAVE_STRIDE_BYTES ((unsigned)TOTAL_WAVES * BYTES_PER_ISSUE) // 9,437,184

// ---------------------------------------------------------------------------
// Kernel 1: stream 604 MB of zeros to d_out using the CDNA5 async LDS->global
// store path (GLOBAL_STORE_ASYNC_FROM_LDS_B128, ASYNCcnt-tracked). Each lane
// sources 16 B from a zeroed LDS staging buffer; TH_STORE_NT avoids polluting
// the 192 MB L2 (output is 604 MB, strictly streaming).
// ---------------------------------------------------------------------------
__global__ __launch_bounds__(FILL_THREADS)
void cb_fill_zero_async(float* __restrict__ out) {
    __shared__ __align__(16) float zbuf[128];  // 512 B of zeros, one 16B slot/lane
    for (int i = threadIdx.x; i < 128; i += FILL_THREADS) zbuf[i] = 0.0f;
    __syncthreads();

    const unsigned lane = threadIdx.x & 31u;
    const unsigned wave = (blockIdx.x * (unsigned)FILL_THREADS + threadIdx.x) >> 5;

    // Low 32 bits of the generic pointer to a __shared__ object == LDS byte
    // offset (aperture bits live in the high word).
    const unsigned lds_off =
        (unsigned)(unsigned long long)(&zbuf[0]) + (lane << 4);

    const unsigned long long base = (unsigned long long)out;  // wave-uniform -> SGPR pair

    // GVS addressing: mem_addr = SADDR(64) + VADDR(u32). 604 MB fits in u32.
    unsigned off = wave * BYTES_PER_ISSUE + (lane << 4);
#pragma unroll 4
    for (unsigned k = 0; k < ISSUES_PER_WAVE; ++k) {
        asm volatile(
            "global_store_async_from_lds_b128 %0, %1, %2 th:TH_STORE_NT"
            :
            : "v"(off), "v"(lds_off), "s"(base)
            : "memory");
        off += WAVE_STRIDE_BYTES;
    }
    // Drain ASYNCcnt (S_ENDPGM would also wait, but be explicit).
    asm volatile("s_wait_asynccnt 0" ::: "memory");
}

// ---------------------------------------------------------------------------
// Kernel 2: scatter the at-most-2304 ones.
//   tid in [0, 2*B*C)      -> gen_poses: coords from sample 0, replicated over b
//   tid in [2*B*C, 4*B*C)  -> step_poses: per-sample interpolated coords
// Semantics faithful to reference:
//   x = int(trunc(cx)) (toward zero), valid iff 0<=x<=255 && 0<=y<=255;
//   step = floor((p2-p1)/3); coords accumulate iteratively (c2 = (p1+s)+s).
//   Invalid points write nothing (clipped write-of-0 is a no-op on zeros).
// ---------------------------------------------------------------------------
__global__ void cb_scatter_ones(const float* __restrict__ p1,
                                const float* __restrict__ p2,
                                float* __restrict__ out) {
    const int tid = blockIdx.x * blockDim.x + threadIdx.x;
    const int NPER = B_ * C_;   // 576
    if (tid >= 4 * NPER) return;

    float cx, cy;
    unsigned long long plane_base;

    if (tid < 2 * NPER) {
        // gen_poses[s, b, c, :, :] from sample-0 coords of pose{s+1}
        const int s = tid / NPER;
        const int r = tid % NPER;
        const int b = r / C_;
        const int c = r % C_;
        const float* p = (s == 0) ? p1 : p2;
        cx = p[2 * c];          // pose[0, c, 0]
        cy = p[2 * c + 1];      // pose[0, c, 1]
        plane_base =
            ((unsigned long long)s * NPER + (unsigned long long)(b * C_ + c)) *
            (unsigned long long)PLANE;
    } else {
        // step_poses[k, b, c, :, :], k in {0,1}
        const int t = tid - 2 * NPER;
        const int k = t / NPER;
        const int r = t % NPER;
        const int b = r / C_;
        const int c = r % C_;
        const int idx = (b * C_ + c) * 2;
        const float p1x = p1[idx], p1y = p1[idx + 1];
        const float p2x = p2[idx], p2y = p2[idx + 1];
        const float sx = floorf((p2x - p1x) / 3.0f);
        const float sy = floorf((p2y - p1y) / 3.0f);
        cx = p1x + sx;
        cy = p1y + sy;
        if (k == 1) { cx += sx; cy += sy; }   // iterative accumulation, FP-faithful
        plane_base = GEN_TOTAL +
                     (unsigned long long)((k * B_ + b) * C_ + c) *
                         (unsigned long long)PLANE;
    }

    const int x = (int)truncf(cx);   // toward zero, like jnp.trunc -> int32
    const int y = (int)truncf(cy);
    if (x >= 0 && x <= 255 && y >= 0 && y <= 255) {
        out[plane_base + (unsigned long long)x * W_ + (unsigned long long)y] = 1.0f;
    }
}

extern "C" void kernel_launch(void* const* d_in, const int* in_sizes, int n_in,
                              void* d_out, int out_size, void* d_ws, size_t ws_size,
                              hipStream_t stream) {
    (void)in_sizes; (void)n_in; (void)out_size; (void)d_ws; (void)ws_size;
    const float* pose1 = (const float*)d_in[0];
    const float* pose2 = (const float*)d_in[1];
    float* out = (float*)d_out;

    // Phase 1: zero all 604 MB of output via async LDS->global B128 stores.
    cb_fill_zero_async<<<FILL_BLOCKS, FILL_THREADS, 0, stream>>>(out);
    // Phase 2 (same stream, ordered): scatter <=2304 ones.
    cb_scatter_ones<<<(4 * B_ * C_ + 255) / 256, 256, 0, stream>>>(pose1, pose2, out);
}